// HybridNN_44727789420657
// MI455X (gfx1250) — compile-verified
//
#include <hip/hip_runtime.h>
#include <hip/hip_bf16.h>
#include <math.h>

// ---------------------------------------------------------------------------
// Problem constants (from the reference)
// ---------------------------------------------------------------------------
#define BATCH     4096
#define IMG_DIM   2048      // 2*32*32
#define HIDDEN    512
#define COM_DIM   128
#define LAYERS    4
#define NQ        12        // ALL_QUBITS
#define NSTATE    4096      // 2^12
#define HALFSTATE 2048

typedef float v2f __attribute__((ext_vector_type(2)));
typedef float v8f __attribute__((ext_vector_type(8)));

// ---------------------------------------------------------------------------
// Weight packing: rearrange W (K x N) into per-(n-tile, k-step) B fragments
// so each wave's B load is one contiguous global_load_b64 per k-step.
//
// Fragment layout (64 floats): Wp[((nt*(K/4) + k4)*64) + lane*2 + v] =
//   W[(4*k4 + (lane>>4)*2 + v) * N + nt*16 + (lane&15)]
// which is exactly the V_WMMA_F32_16X16X4_F32 B-operand layout:
//   lanes 0-15 : v0=row K+0, v1=row K+1 (col = lane)
//   lanes 16-31: v0=row K+2, v1=row K+3 (col = lane-16)
// ---------------------------------------------------------------------------
__global__ __launch_bounds__(256)
void pack_w_kernel(const float* __restrict__ W, float* __restrict__ Wp,
                   int K, int N)
{
    int idx = blockIdx.x * 256 + threadIdx.x;
    if (idx >= K * N) return;
    int frag = idx >> 6;          // 64-float fragment id
    int off  = idx & 63;
    int lane = off >> 1;
    int v    = off & 1;
    int K4   = K >> 2;
    int k4   = frag % K4;
    int nt   = frag / K4;
    int krow = (k4 << 2) + ((lane >> 4) << 1) + v;
    int col  = (nt << 4) + (lane & 15);
    Wp[idx] = W[(size_t)krow * N + col];
}

// ---------------------------------------------------------------------------
// fp32 GEMM via V_WMMA_F32_16X16X4_F32, one wave per 16x16 tile, 8 waves/blk.
// N, K compile-time -> all inner-loop addresses are base + immediate offset.
// 1-deep software pipeline so loads for k4+1 overlap the WMMA of k4.
//
// A-fragment (ISA 7.12.2, 32-bit A 16x4):
//   lanes 0-15 : M=lane,    v0=K+0, v1=K+1   (contiguous 8B load)
//   lanes 16-31: M=lane-16, v0=K+2, v1=K+3
// C/D: VGPR r -> M = r (+8 for lanes 16-31), N = lane&15.
// ---------------------------------------------------------------------------
template<int N, int K, int RELU>
__global__ __launch_bounds__(256)
void gemm_wmma_f32(const float* __restrict__ A,    // M x K
                   const float* __restrict__ Wp,   // packed weights
                   const float* __restrict__ bias, // N
                   float* __restrict__ C)          // M x N
{
    constexpr int K4     = K / 4;
    constexpr int tilesN = N / 16;

    const int lane = threadIdx.x & 31;
    const int wave = threadIdx.x >> 5;
    const int tile = blockIdx.x * 8 + wave;

    const int m0 = (tile / tilesN) << 4;
    const int n0 = (tile % tilesN) << 4;

    const int halfsel = lane >> 4;          // 0 or 1
    const int koff    = halfsel << 1;       // 0 or 2
    const int lrow    = lane & 15;

    // A: per-lane row pointer; element k4 at immediate offset k4*16 bytes
    const float2* Ap = (const float2*)(A + (size_t)(m0 + lrow) * K + koff);
    // B: contiguous fragment stream; element k4 at immediate offset k4*256 B
    const float2* Bp = (const float2*)(Wp + (size_t)(n0 >> 4) * K4 * 64) + lane;

    v8f acc = {};
    v2f a, b;
    { float2 t = Ap[0]; a.x = t.x; a.y = t.y; }
    { float2 t = Bp[0]; b.x = t.x; b.y = t.y; }

#pragma unroll 8
    for (int k4 = 0; k4 < K4 - 1; ++k4) {
        float2 ta = Ap[(k4 + 1) * 2];
        float2 tb = Bp[(size_t)(k4 + 1) * 32];
        acc = __builtin_amdgcn_wmma_f32_16x16x4_f32(
                  false, a, false, b, (short)0, acc, false, false);
        a.x = ta.x; a.y = ta.y;
        b.x = tb.x; b.y = tb.y;
    }
    acc = __builtin_amdgcn_wmma_f32_16x16x4_f32(
              false, a, false, b, (short)0, acc, false, false);

    const int col = n0 + lrow;
    const float bv = bias[col];
#pragma unroll
    for (int r = 0; r < 8; ++r) {
        int row = m0 + r + (halfsel << 3);
        float v = acc[r] + bv;
        if (RELU) v = fmaxf(v, 0.0f);
        C[(size_t)row * N + col] = v;
    }
}

// ---------------------------------------------------------------------------
// 12-qubit state-vector circuit, one block per batch element, state in LDS.
// Qubit w lives at bit (11-w) of the flat index (stride 2^(11-w)).
// ---------------------------------------------------------------------------
__device__ __forceinline__ void h_gate(float* re, float* im, int w, int tid)
{
    const int st = 1 << (11 - w);
    const float r = 0.70710678118654752440f;
    for (int p = tid; p < HALFSTATE; p += 256) {
        int i0 = ((p & ~(st - 1)) << 1) | (p & (st - 1));
        int i1 = i0 + st;
        float a0r = re[i0], a0i = im[i0], a1r = re[i1], a1i = im[i1];
        re[i0] = (a0r + a1r) * r;  im[i0] = (a0i + a1i) * r;
        re[i1] = (a0r - a1r) * r;  im[i1] = (a0i - a1i) * r;
    }
    __syncthreads();
}

__device__ __forceinline__ void rot_gate(float* re, float* im, int w,
                                         float phi, float theta, float omega,
                                         int tid)
{
    float st_, ct, sb, cb, sa, ca;
    __sincosf(0.5f * theta,         &st_, &ct);
    __sincosf(0.5f * (phi + omega), &sb,  &cb);
    __sincosf(0.5f * (phi - omega), &sa,  &ca);
    // m00 = e^{-i(phi+omega)/2} c ; m01 = -e^{+i(phi-omega)/2} s
    // m10 = e^{-i(phi-omega)/2} s ; m11 = e^{+i(phi+omega)/2} c
    const float m00r =  cb * ct,  m00i = -sb * ct;
    const float m01r = -ca * st_, m01i = -sa * st_;
    const float m10r =  ca * st_, m10i = -sa * st_;
    const float m11r =  cb * ct,  m11i =  sb * ct;

    const int st = 1 << (11 - w);
    for (int p = tid; p < HALFSTATE; p += 256) {
        int i0 = ((p & ~(st - 1)) << 1) | (p & (st - 1));
        int i1 = i0 + st;
        float a0r = re[i0], a0i = im[i0], a1r = re[i1], a1i = im[i1];
        float n0r = m00r*a0r - m00i*a0i + m01r*a1r - m01i*a1i;
        float n0i = m00r*a0i + m00i*a0r + m01r*a1i + m01i*a1r;
        float n1r = m10r*a0r - m10i*a0i + m11r*a1r - m11i*a1i;
        float n1i = m10r*a0i + m10i*a0r + m11r*a1i + m11i*a1r;
        re[i0] = n0r; im[i0] = n0i;
        re[i1] = n1r; im[i1] = n1i;
    }
    __syncthreads();
}

__global__ __launch_bounds__(256)
void circuit_kernel(const float* __restrict__ x,    // BATCH x IMG_DIM (img_params = -x)
                    const float* __restrict__ com,  // BATCH x COM_DIM
                    const float* __restrict__ asz,  // LAYERS x 12 x 3
                    float* __restrict__ out)        // BATCH
{
    __shared__ float sre[NSTATE];
    __shared__ float sim_[NSTATE];
    __shared__ float cA[COM_DIM];
    __shared__ float sA[COM_DIM];
    __shared__ float red[8];

    const int b   = blockIdx.x;
    const int tid = threadIdx.x;

    // |0...0>
    for (int i = tid; i < NSTATE; i += 256) { sre[i] = 0.0f; sim_[i] = 0.0f; }
    if (tid == 0) sre[0] = 1.0f;

    // Precompute FRQI(com) cos/sin with bit-reversed angle index (q=7).
    if (tid < COM_DIM) {
        unsigned rj = __brev((unsigned)tid) >> 25;           // bitrev7
        float a = 0.5f * com[(size_t)b * COM_DIM + rj];
        float s, c; __sincosf(a, &s, &c);
        cA[tid] = c; sA[tid] = s;
    }
    __syncthreads();

    // H on wires 1..7
    for (int w = 1; w <= 7; ++w) h_gate(sre, sim_, w, tid);

    // FRQI(com, q=7): RY(ang[j]) on qubit 0, j = bits 10..4 of index
    for (int p = tid; p < HALFSTATE; p += 256) {
        int j = (p >> 4) & 127;
        float c = cA[j], s = sA[j];
        int i1 = p + HALFSTATE;
        float a0r = sre[p], a0i = sim_[p], a1r = sre[i1], a1i = sim_[i1];
        sre[p]  = c*a0r - s*a1r;  sim_[p]  = c*a0i - s*a1i;
        sre[i1] = s*a0r + c*a1r;  sim_[i1] = s*a0i + c*a1i;
    }
    __syncthreads();

    // Variational layers
    for (int l = 0; l < LAYERS; ++l) {
        for (int w = 0; w < NQ; ++w) {
            const float* p3 = asz + ((size_t)l * NQ + w) * 3;
            rot_gate(sre, sim_, w, p3[0], p3[1], p3[2], tid);
        }
        // CNOT ring composed into a single permutation gather:
        // state'[i] = state[G_w0(G_w1(...G_w11(i)))], G applied w=11..0.
        const int r = l % (NQ - 1) + 1;
        float tr[16], ti[16];
#pragma unroll
        for (int t = 0; t < 16; ++t) {
            int i = tid + t * 256;
            int idx = i;
#pragma unroll
            for (int w = NQ - 1; w >= 0; --w) {
                int csh = 11 - w;
                int tsh = 11 - ((w + r) % NQ);
                idx ^= ((idx >> csh) & 1) << tsh;
            }
            tr[t] = sre[idx]; ti[t] = sim_[idx];
        }
        __syncthreads();
#pragma unroll
        for (int t = 0; t < 16; ++t) {
            int i = tid + t * 256;
            sre[i] = tr[t]; sim_[i] = ti[t];
        }
        __syncthreads();
    }

    // FRQI(img = -x, q=11): angle index = bitrev11(pair index), rest=1
    for (int p = tid; p < HALFSTATE; p += 256) {
        unsigned rp = __brev((unsigned)p) >> 21;             // bitrev11
        float a = -0.5f * x[(size_t)b * IMG_DIM + rp];
        float s, c; __sincosf(a, &s, &c);
        int i1 = p + HALFSTATE;
        float a0r = sre[p], a0i = sim_[p], a1r = sre[i1], a1i = sim_[i1];
        sre[p]  = c*a0r - s*a1r;  sim_[p]  = c*a0i - s*a1i;
        sre[i1] = s*a0r + c*a1r;  sim_[i1] = s*a0i + c*a1i;
    }
    __syncthreads();

    // H on wires 1..11
    for (int w = 1; w <= 11; ++w) h_gate(sre, sim_, w, tid);

    // out[b] = P(q0=1) - P(q0=0)
    float acc = 0.0f;
    for (int i = tid; i < NSTATE; i += 256) {
        float v = sre[i]*sre[i] + sim_[i]*sim_[i];
        acc += (i & HALFSTATE) ? v : -v;
    }
    for (int off = 16; off > 0; off >>= 1)
        acc += __shfl_down(acc, off, 32);
    if ((tid & 31) == 0) red[tid >> 5] = acc;
    __syncthreads();
    if (tid == 0) {
        float s = 0.0f;
#pragma unroll
        for (int i = 0; i < 8; ++i) s += red[i];
        out[b] = s;
    }
}

// ---------------------------------------------------------------------------
// Launcher
// Inputs: 0=x (B,2,32,32), 1=W1 (2048,512), 2=b1 (512), 3=W2 (512,128),
//         4=b2 (128), 5=asz_params (4,12,3). Output: (4096,) f32.
// Workspace: H1 8MB | com 2MB | packed W1 4MB | packed W2 256KB
// ---------------------------------------------------------------------------
extern "C" void kernel_launch(void* const* d_in, const int* in_sizes, int n_in,
                              void* d_out, int out_size, void* d_ws, size_t ws_size,
                              hipStream_t stream)
{
    (void)in_sizes; (void)n_in; (void)out_size; (void)ws_size;
    const float* x   = (const float*)d_in[0];
    const float* W1  = (const float*)d_in[1];
    const float* b1  = (const float*)d_in[2];
    const float* W2  = (const float*)d_in[3];
    const float* b2  = (const float*)d_in[4];
    const float* asz = (const float*)d_in[5];
    float* out = (float*)d_out;

    float* H1  = (float*)d_ws;                        // 4096 x 512
    float* com = H1  + (size_t)BATCH * HIDDEN;        // 4096 x 128
    float* Wp1 = com + (size_t)BATCH * COM_DIM;       // 2048 x 512 packed
    float* Wp2 = Wp1 + (size_t)IMG_DIM * HIDDEN;      // 512 x 128 packed

    // Pack weights into WMMA B-fragment layout (contiguous b64 per lane/step)
    pack_w_kernel<<<(IMG_DIM * HIDDEN) / 256, 256, 0, stream>>>(W1, Wp1, IMG_DIM, HIDDEN);
    pack_w_kernel<<<(HIDDEN * COM_DIM) / 256, 256, 0, stream>>>(W2, Wp2, HIDDEN, COM_DIM);

    // GEMM1 + ReLU: H1 = relu(x @ W1 + b1)   (4096x2048x512)
    gemm_wmma_f32<HIDDEN, IMG_DIM, 1>
        <<<(BATCH / 16) * (HIDDEN / 16) / 8, 256, 0, stream>>>(x, Wp1, b1, H1);

    // GEMM2: com = H1 @ W2 + b2              (4096x512x128)
    gemm_wmma_f32<COM_DIM, HIDDEN, 0>
        <<<(BATCH / 16) * (COM_DIM / 16) / 8, 256, 0, stream>>>(H1, Wp2, b2, com);

    // Quantum circuit: one block per batch element, state in LDS
    circuit_kernel<<<BATCH, 256, 0, stream>>>(x, com, asz, out);
}